// MultiHeadAttention_23244363006423
// MI455X (gfx1250) — compile-verified
//
#include <hip/hip_runtime.h>
#include <hip/hip_bf16.h>

// MHA: B=4, S=2048, F=64, H=8, D=8.
// Outputs: [0,256) = sum-over-seq projected output [4,64]
//          [256, 256+8*4*2048*2048) = softmax weights [H,B,S,S]

#define S_LEN 2048
#define HB_N 32          // H*B
#define DEPTH 8
#define STRIPS 128       // S/16

typedef float v2f __attribute__((ext_vector_type(2)));
typedef float v4f __attribute__((ext_vector_type(4)));
typedef float v8f __attribute__((ext_vector_type(8)));

// ---------------------------------------------------------------------------
// Kernel A: q/k/v projections. One block = 4 tokens x 64 (h,d) outputs.
// q,v layout: [hb][s][d] ; k layout transposed: [hb][d][s] (for coalesced
// WMMA B-fragment loads). Block 0 also zeroes the head-sum accumulator.
// ---------------------------------------------------------------------------
__global__ __launch_bounds__(256) void proj_kernel(
    const float* __restrict__ x,
    const float* __restrict__ Wq, const float* __restrict__ Wk,
    const float* __restrict__ Wv,
    float* __restrict__ q, float* __restrict__ kT, float* __restrict__ v,
    float* __restrict__ headsum)
{
    __shared__ float xs[4 * 64];
    __shared__ float lwq[64 * 64];
    __shared__ float lwk[64 * 64];
    __shared__ float lwv[64 * 64];

    const int tid = threadIdx.x;
    if (blockIdx.x == 0) headsum[tid] = 0.0f;  // 256 entries == blockDim

    // Weights [H][F][D] -> LDS as [f][h*8+d] (conflict-free column reads)
    #pragma unroll
    for (int i = 0; i < 16; ++i) {
        int idx = i * 256 + tid;            // 0..4095
        int h = idx >> 9;
        int f = (idx >> 3) & 63;
        int d = idx & 7;
        int dst = f * 64 + h * 8 + d;
        lwq[dst] = Wq[idx];
        lwk[dst] = Wk[idx];
        lwv[dst] = Wv[idx];
    }
    const int t0 = blockIdx.x * 4;
    xs[tid] = x[(size_t)t0 * 64 + tid];     // 4 tokens x 64 feats
    __syncthreads();

    const int j  = tid >> 6;                // token within block
    const int hd = tid & 63;
    float aq = 0.f, ak = 0.f, av = 0.f;
    #pragma unroll 8
    for (int f = 0; f < 64; ++f) {
        float xv = xs[j * 64 + f];
        aq = fmaf(xv, lwq[f * 64 + hd], aq);
        ak = fmaf(xv, lwk[f * 64 + hd], ak);
        av = fmaf(xv, lwv[f * 64 + hd], av);
    }
    const int token = t0 + j;
    const int b = token >> 11;
    const int s = token & (S_LEN - 1);
    const int h = hd >> 3;
    const int d = hd & 7;
    const int hb = h * 4 + b;
    q [((size_t)hb * S_LEN + s) * DEPTH + d] = aq;
    kT[((size_t)hb * DEPTH + d) * S_LEN + s] = ak;
    v [((size_t)hb * S_LEN + s) * DEPTH + d] = av;
}

// ---------------------------------------------------------------------------
// Kernel B: one block per (hb, 16-row strip). 128 threads = 4 waves.
// Scores via v_wmma_f32_16x16x4_f32 (K=8 -> two WMMAs), softmax in LDS,
// single coalesced non-temporal b128 write of the weights + head-sum acc.
// Dynamic LDS: 16*2048 scores + reductions (~129 KB; WGP has 320 KB).
// ---------------------------------------------------------------------------
__global__ __launch_bounds__(128) void attn_kernel(
    const float* __restrict__ q, const float* __restrict__ kT,
    const float* __restrict__ v,
    float* __restrict__ out_w, float* __restrict__ headsum)
{
    extern __shared__ float smem[];
    float* sc     = smem;                 // [16][2048]
    float* red    = smem + 16 * S_LEN;    // [16][8]
    float* rowmax = red + 128;            // [16]
    float* rowinv = rowmax + 16;          // [16]
    float* part   = rowinv + 16;          // [8]

    const int tid   = threadIdx.x;
    const int lane  = tid & 31;
    const int wave  = tid >> 5;
    const int strip = blockIdx.x & (STRIPS - 1);
    const int hb    = blockIdx.x >> 7;

    const int mrow  = lane & 15;          // M (row in tile) / N (col in tile)
    const int khalf = lane >> 4;          // lane half selects K pairs

    // A fragment: q rows of this strip. ISA A-layout 16x4 f32:
    // VGPR j holds K=j (lanes 0-15) / K=j+2 (lanes 16-31).
    const float* qb = q + ((size_t)hb * S_LEN + strip * 16 + mrow) * DEPTH;
    v2f a0, a1;
    a0.x = qb[2 * khalf + 0];
    a0.y = qb[2 * khalf + 1];
    a1.x = qb[4 + 2 * khalf + 0];
    a1.y = qb[4 + 2 * khalf + 1];

    const float* kb = kT + (size_t)hb * DEPTH * S_LEN;
    const float scale = 0.35355339059327373f;  // 1/sqrt(8)

    for (int tile = wave; tile < STRIPS; tile += 4) {
        const int tcol = tile * 16 + mrow;
        v2f b0, b1;
        b0.x = kb[(size_t)(2 * khalf + 0) * S_LEN + tcol];
        b0.y = kb[(size_t)(2 * khalf + 1) * S_LEN + tcol];
        b1.x = kb[(size_t)(4 + 2 * khalf + 0) * S_LEN + tcol];
        b1.y = kb[(size_t)(4 + 2 * khalf + 1) * S_LEN + tcol];

        v8f acc = {0.f, 0.f, 0.f, 0.f, 0.f, 0.f, 0.f, 0.f};
        acc = __builtin_amdgcn_wmma_f32_16x16x4_f32(
            false, a0, false, b0, (short)0, acc, false, false);
        acc = __builtin_amdgcn_wmma_f32_16x16x4_f32(
            false, a1, false, b1, (short)0, acc, false, false);

        // C/D layout: VGPR r holds M=r (lanes 0-15) / M=r+8 (lanes 16-31)
        #pragma unroll
        for (int r = 0; r < 8; ++r) {
            int m = r + 8 * khalf;
            sc[m * S_LEN + tile * 16 + mrow] = acc[r] * scale;
        }
    }
    __syncthreads();

    // ---- row softmax over 2048: 8 threads per row, b128 chunks ----
    const int r = tid >> 3;    // 0..15
    const int g = tid & 7;     // 0..7
    v4f* sc4 = (v4f*)sc;
    const int base4 = r * (S_LEN / 4) + g;     // v4f index

    float mx = -3.0e38f;
    for (int j2 = 0; j2 < 64; ++j2) {
        v4f t = sc4[base4 + 8 * j2];           // ds_load_b128
        mx = fmaxf(mx, fmaxf(fmaxf(t.x, t.y), fmaxf(t.z, t.w)));
    }
    red[r * 8 + g] = mx;
    __syncthreads();
    if (g == 0) {
        float mm = red[r * 8];
        #pragma unroll
        for (int i = 1; i < 8; ++i) mm = fmaxf(mm, red[r * 8 + i]);
        rowmax[r] = mm;
    }
    __syncthreads();
    const float rm = rowmax[r];
    float sum = 0.f;
    for (int j2 = 0; j2 < 64; ++j2) {
        v4f t = sc4[base4 + 8 * j2];
        t.x = __expf(t.x - rm);
        t.y = __expf(t.y - rm);
        t.z = __expf(t.z - rm);
        t.w = __expf(t.w - rm);
        sc4[base4 + 8 * j2] = t;               // ds_store_b128
        sum += (t.x + t.y) + (t.z + t.w);
    }
    red[r * 8 + g] = sum;
    __syncthreads();
    if (g == 0) {
        float ss = 0.f;
        #pragma unroll
        for (int i = 0; i < 8; ++i) ss += red[r * 8 + i];
        rowinv[r] = 1.0f / ss;
    }
    if (tid < 8) part[tid] = 0.0f;
    __syncthreads();

    // ---- coalesced non-temporal b128 weight store + head-sum partials ----
    float accv[8] = {0.f, 0.f, 0.f, 0.f, 0.f, 0.f, 0.f, 0.f};
    float* wb = out_w + (size_t)hb * S_LEN * S_LEN + (size_t)(strip * 16) * S_LEN;
    const v4f* vb4 = (const v4f*)(v + (size_t)hb * S_LEN * DEPTH);
    for (int it = 0; it < 64; ++it) {
        int idx = it * 512 + tid * 4;          // linear float idx over [16][2048]
        int rr  = idx >> 11;
        int cc  = idx & (S_LEN - 1);
        float inv = rowinv[rr];
        v4f sv = *(const v4f*)(sc + idx);
        v4f wv = sv * inv;
        __builtin_nontemporal_store(wv, (v4f*)(wb + (size_t)rr * S_LEN + cc));
        #pragma unroll
        for (int u = 0; u < 4; ++u) {
            float w = wv[u];
            v4f v0 = vb4[(size_t)(cc + u) * 2 + 0];
            v4f v1 = vb4[(size_t)(cc + u) * 2 + 1];
            accv[0] = fmaf(w, v0.x, accv[0]);
            accv[1] = fmaf(w, v0.y, accv[1]);
            accv[2] = fmaf(w, v0.z, accv[2]);
            accv[3] = fmaf(w, v0.w, accv[3]);
            accv[4] = fmaf(w, v1.x, accv[4]);
            accv[5] = fmaf(w, v1.y, accv[5]);
            accv[6] = fmaf(w, v1.z, accv[6]);
            accv[7] = fmaf(w, v1.w, accv[7]);
        }
    }
    #pragma unroll
    for (int d = 0; d < DEPTH; ++d) atomicAdd(&part[d], accv[d]);
    __syncthreads();
    if (tid < 8) atomicAdd(&headsum[hb * DEPTH + tid], part[tid]);
}

// ---------------------------------------------------------------------------
// Kernel C: out[b,f] = sum_{h,d} headsum[h*4+b][d] * Wo[h*8+d][f]
// ---------------------------------------------------------------------------
__global__ __launch_bounds__(256) void finalize_kernel(
    const float* __restrict__ headsum, const float* __restrict__ Wo,
    float* __restrict__ out_sum)
{
    const int tid = threadIdx.x;
    const int b = tid >> 6;
    const int f = tid & 63;
    float acc = 0.f;
    #pragma unroll 8
    for (int hd = 0; hd < 64; ++hd) {
        int h = hd >> 3, d = hd & 7;
        acc = fmaf(headsum[(h * 4 + b) * DEPTH + d], Wo[hd * 64 + f], acc);
    }
    out_sum[b * 64 + f] = acc;
}

// ---------------------------------------------------------------------------
extern "C" void kernel_launch(void* const* d_in, const int* in_sizes, int n_in,
                              void* d_out, int out_size, void* d_ws, size_t ws_size,
                              hipStream_t stream) {
    const float* x  = (const float*)d_in[0];
    const float* Wq = (const float*)d_in[1];
    const float* Wk = (const float*)d_in[2];
    const float* Wv = (const float*)d_in[3];
    const float* Wo = (const float*)d_in[4];

    // Workspace: q (2MB) | kT (2MB) | v (2MB) | headsum (1KB)  (~6.3MB)
    float* ws = (float*)d_ws;
    float* q  = ws;
    float* kT = ws + (size_t)HB_N * S_LEN * DEPTH;      // +524288
    float* v  = kT + (size_t)HB_N * S_LEN * DEPTH;      // +524288
    float* hs = v  + (size_t)HB_N * S_LEN * DEPTH;      // +524288

    float* out_sum = (float*)d_out;           // [4,64]
    float* out_w   = out_sum + 256;           // [8,4,2048,2048]

    const size_t smem = (size_t)(16 * S_LEN + 128 + 16 + 16 + 8) * sizeof(float);
    (void)hipFuncSetAttribute((const void*)attn_kernel,
                              hipFuncAttributeMaxDynamicSharedMemorySize,
                              (int)smem);

    proj_kernel<<<2048, 256, 0, stream>>>(x, Wq, Wk, Wv, q, kT, v, hs);
    attn_kernel<<<HB_N * STRIPS, 128, smem, stream>>>(q, kT, v, out_w, hs);
    finalize_kernel<<<1, 256, 0, stream>>>(hs, Wo, out_sum);
}